// SurpriseGatedDeltaRecurrence_47467978555543
// MI455X (gfx1250) — compile-verified
//
#include <hip/hip_runtime.h>
#include <hip/hip_bf16.h>

// Surprise-gated delta recurrence for MI455X (gfx1250).
// fp32 throughout; matrix work on V_WMMA_F32_16X16X4_F32.
// B=4, S=2048, D_MODEL=2048, D_STATE=64.

typedef __attribute__((ext_vector_type(2))) float v2f;
typedef __attribute__((ext_vector_type(8))) float v8f;

#define DSTATE 64
#define DMODEL 2048
#define BATCH  4
#define SEQ    2048
#define MROWS  (BATCH * SEQ)   // 8192 rows of x / projections

static __device__ __forceinline__ v8f wmma_f32(v2f a, v2f b, v8f c) {
    // D = A(16x4, f32) * B(4x16, f32) + C(16x16, f32)
    return __builtin_amdgcn_wmma_f32_16x16x4_f32(
        /*neg_a=*/false, a, /*neg_b=*/false, b,
        /*c_mod=*/(short)0, c, /*reuse_a=*/false, /*reuse_b=*/false);
}

static __device__ __forceinline__ float sigmoidf(float z) {
    return 1.0f / (1.0f + expf(-z));
}

// ---------------------------------------------------------------------------
// Kernel 1: fused projection GEMM.
// grid.x = 512 (16-row tiles of x); block = 128 threads = 4 waves;
// wave w computes projection w's full 16x64 tile with 4 accumulators.
// Per K-step of 4: 1 A load (float2) feeds 4 WMMAs (one per 16-col tile).
// A layout (16x4 f32): lane&15 = M row; lane>>4 selects K pair {2h,2h+1}.
// B layout (4x16 f32): lane&15 = N col; same K-pair mapping.
// ---------------------------------------------------------------------------
__global__ __launch_bounds__(128) void proj_gemm_kernel(
    const float* __restrict__ x,     // (MROWS, DMODEL)
    const float* __restrict__ Wk,    // (DSTATE, DMODEL)
    const float* __restrict__ Wv,
    const float* __restrict__ Wq,
    const float* __restrict__ Wa,
    float* __restrict__ kraw,        // (MROWS, DSTATE)
    float* __restrict__ vraw,
    float* __restrict__ qraw,
    float* __restrict__ araw)
{
    const int mt   = blockIdx.x;            // 0..511 : 16-row tile of x
    const int wave = threadIdx.x >> 5;      // 0..3   : projection index
    const int lane = threadIdx.x & 31;

    const float* W   = (wave == 0) ? Wk : (wave == 1) ? Wv : (wave == 2) ? Wq : Wa;
    float*       Out = (wave == 0) ? kraw : (wave == 1) ? vraw : (wave == 2) ? qraw : araw;

    const int row  = mt * 16 + (lane & 15);
    const int coln = lane & 15;             // col within a 16-wide N tile
    const int koff = (lane >> 4) * 2;       // K pair for this lane half

    const float* ap  = x + (size_t)row * DMODEL + koff;
    const float* bp0 = W + (size_t)(0 * 16 + coln) * DMODEL + koff;
    const float* bp1 = W + (size_t)(1 * 16 + coln) * DMODEL + koff;
    const float* bp2 = W + (size_t)(2 * 16 + coln) * DMODEL + koff;
    const float* bp3 = W + (size_t)(3 * 16 + coln) * DMODEL + koff;

    v8f acc0 = {}, acc1 = {}, acc2 = {}, acc3 = {};
    for (int kk = 0; kk < DMODEL; kk += 16) {
#pragma unroll
        for (int u = 0; u < 4; ++u) {
            const int o = kk + 4 * u;
            v2f a  = *(const v2f*)(ap  + o);
            v2f b0 = *(const v2f*)(bp0 + o);
            v2f b1 = *(const v2f*)(bp1 + o);
            v2f b2 = *(const v2f*)(bp2 + o);
            v2f b3 = *(const v2f*)(bp3 + o);
            acc0 = wmma_f32(a, b0, acc0);
            acc1 = wmma_f32(a, b1, acc1);
            acc2 = wmma_f32(a, b2, acc2);
            acc3 = wmma_f32(a, b3, acc3);
        }
    }

    // C/D layout: VGPR i -> M = i (lanes 0-15) or M = 8+i (lanes 16-31)
    const int r0 = mt * 16 + ((lane >> 4) ? 8 : 0);
#pragma unroll
    for (int i = 0; i < 8; ++i) {
        Out[(size_t)(r0 + i) * DSTATE + 0  + coln] = acc0[i];
        Out[(size_t)(r0 + i) * DSTATE + 16 + coln] = acc1[i];
        Out[(size_t)(r0 + i) * DSTATE + 32 + coln] = acc2[i];
        Out[(size_t)(r0 + i) * DSTATE + 48 + coln] = acc3[i];
    }
}

// ---------------------------------------------------------------------------
// Kernel 2: per-row ops, in place.  One wave per row (2 elems/lane, wave32).
//   k <- l2norm(k); q <- l2norm(q); araw <- alpha = exp(C * sigmoid(.) * log_a)
// ---------------------------------------------------------------------------
__global__ __launch_bounds__(256) void row_ops_kernel(
    float* __restrict__ kraw,
    float* __restrict__ qraw,
    float* __restrict__ araw,
    const float* __restrict__ Wa_b,   // (64,)
    const float* __restrict__ lam)    // (64,)
{
    const int wave = threadIdx.x >> 5;
    const int lane = threadIdx.x & 31;
    const int row  = blockIdx.x * 8 + wave;
    const int e0   = lane * 2;

    // --- l2norm(k) ---
    {
        v2f kv = *(const v2f*)(kraw + (size_t)row * DSTATE + e0);
        float ss = kv.x * kv.x + kv.y * kv.y;
#pragma unroll
        for (int m = 16; m >= 1; m >>= 1) ss += __shfl_xor(ss, m, 32);
        float scale = 1.0f / fmaxf(sqrtf(ss), 1e-12f);
        kv.x *= scale; kv.y *= scale;
        *(v2f*)(kraw + (size_t)row * DSTATE + e0) = kv;
    }
    // --- l2norm(q) ---
    {
        v2f qv = *(const v2f*)(qraw + (size_t)row * DSTATE + e0);
        float ss = qv.x * qv.x + qv.y * qv.y;
#pragma unroll
        for (int m = 16; m >= 1; m >>= 1) ss += __shfl_xor(ss, m, 32);
        float scale = 1.0f / fmaxf(sqrtf(ss), 1e-12f);
        qv.x *= scale; qv.y *= scale;
        *(v2f*)(qraw + (size_t)row * DSTATE + e0) = qv;
    }
    // --- alpha = exp(C * sigmoid(raw + b) * log(sigmoid(lam) + 1e-8)) ---
    {
        v2f rv = *(const v2f*)(araw + (size_t)row * DSTATE + e0);
        float r0 = sigmoidf(rv.x + Wa_b[e0]);
        float r1 = sigmoidf(rv.y + Wa_b[e0 + 1]);
        float la0 = logf(sigmoidf(lam[e0])     + 1e-8f);
        float la1 = logf(sigmoidf(lam[e0 + 1]) + 1e-8f);
        v2f av;
        av.x = expf(8.0f * r0 * la0);
        av.y = expf(8.0f * r1 * la1);
        *(v2f*)(araw + (size_t)row * DSTATE + e0) = av;
    }
}

// ---------------------------------------------------------------------------
// Kernel 3: the sequential scan.  One block per batch (grid=4), 256 threads.
// Thread tid owns H[k][v] for v = tid&63 and k in [16*(tid>>6), +16),
// kept in 16 registers.  Fused RMSNorm epilogue writes yn directly.
// ---------------------------------------------------------------------------
__global__ __launch_bounds__(256) void scan_kernel(
    const float* __restrict__ kbuf,   // (B, S, 64) normalized
    const float* __restrict__ vbuf,
    const float* __restrict__ qbuf,
    const float* __restrict__ abuf,   // alpha
    const float* __restrict__ norm_w, // (64,)
    float* __restrict__ ynbuf)        // (B, S, 64) rmsnormed output
{
    const int b   = blockIdx.x;
    const int tid = threadIdx.x;
    const int v   = tid & 63;         // my column of H
    const int kc  = tid >> 6;         // my 16-row k block (0..3)

    __shared__ float sk[64], sv[64], sq[64], sa[64];
    __shared__ float p1[256], p2[256];
    __shared__ float su[64], sy[64];
    __shared__ float wred[8];
    __shared__ float s_surprise, s_scale;

    float h[16];
#pragma unroll
    for (int i = 0; i < 16; ++i) h[i] = 0.0f;

    // this thread's streaming source: element v of one of the four tensors
    const float* src =
        (kc == 0) ? kbuf : (kc == 1) ? vbuf : (kc == 2) ? qbuf : abuf;
    src += (size_t)b * SEQ * DSTATE + v;
    float* dstS = (kc == 0) ? sk : (kc == 1) ? sv : (kc == 2) ? sq : sa;

    float*       yout = ynbuf + (size_t)b * SEQ * DSTATE;
    const float  nw   = (tid < 64) ? norm_w[v] : 0.0f;

    for (int t = 0; t < SEQ; ++t) {
        // ---- stage k_t, v_t, q_t, a_t into LDS (1 load / thread) ----
        dstS[v] = src[(size_t)t * DSTATE];
        if (t + 8 < SEQ)
            __builtin_prefetch(src + (size_t)(t + 8) * DSTATE, 0, 0);
        __syncthreads();

        // ---- partial matvecs: pred = q.H, kH = k.H (pre-update H) ----
        float pp = 0.0f, pk = 0.0f;
#pragma unroll
        for (int j = 0; j < 16; ++j) {
            const int kk = kc * 16 + j;
            pp = fmaf(sq[kk], h[j], pp);
            pk = fmaf(sk[kk], h[j], pk);
        }
        p1[tid] = pp;
        p2[tid] = pk;
        __syncthreads();

        // ---- finish matvecs, err partial, u = v - kH ----
        float err_part = 0.0f;
        if (tid < 64) {
            const float pred = p1[v] + p1[v + 64] + p1[v + 128] + p1[v + 192];
            const float kH   = p2[v] + p2[v + 64] + p2[v + 128] + p2[v + 192];
            su[v] = sv[v] - kH;
            const float d = sv[v] - pred;
            err_part = d * d;
        }
        float e = err_part;
#pragma unroll
        for (int m = 16; m >= 1; m >>= 1) e += __shfl_xor(e, m, 32);
        if ((tid & 31) == 0 && tid < 64) wred[tid >> 5] = e;
        __syncthreads();

        if (tid == 0) {
            const float err = wred[0] + wred[1];
            s_surprise = sigmoidf(err * (1.0f / 1.000001f)); // / (TEMP + 1e-6)
        }
        __syncthreads();

        // ---- H update + y partial:  H = a*H + (1-a)*s*(k (x) u) ----
        const float sp = s_surprise;
        const float uu = sp * su[v];       // s * u[v], shared factor
        float yp = 0.0f;
#pragma unroll
        for (int j = 0; j < 16; ++j) {
            const int kk = kc * 16 + j;
            const float a = sa[kk];
            h[j] = a * h[j] + (1.0f - a) * (sk[kk] * uu);
            yp = fmaf(sq[kk], h[j], yp);
        }
        p1[tid] = yp;
        __syncthreads();

        // ---- finish y, RMS partial ----
        float ms_part = 0.0f;
        if (tid < 64) {
            const float yv = p1[v] + p1[v + 64] + p1[v + 128] + p1[v + 192];
            sy[v] = yv;
            ms_part = yv * yv;
        }
        float m2 = ms_part;
#pragma unroll
        for (int m = 16; m >= 1; m >>= 1) m2 += __shfl_xor(m2, m, 32);
        if ((tid & 31) == 0 && tid < 64) wred[tid >> 5] = m2;
        __syncthreads();

        if (tid == 0)
            s_scale = rsqrtf((wred[0] + wred[1]) * (1.0f / 64.0f) + 1e-6f);
        __syncthreads();

        // ---- fused RMSNorm epilogue -> yn ----
        if (tid < 64)
            yout[(size_t)t * DSTATE + v] = sy[v] * s_scale * nw;
        // next iteration's LDS writes are fenced by its own __syncthreads()
    }
}

// ---------------------------------------------------------------------------
// Kernel 4: output GEMM  out = yn @ Wo^T,  (8192x64)x(64x2048), K=64.
// Each wave keeps the whole A strip (16 rows x K=64 -> 16 float2/lane) in
// registers and runs 4 N-tiles against it: 80 loads per 64 WMMAs.
// ---------------------------------------------------------------------------
__global__ __launch_bounds__(128) void out_gemm_kernel(
    const float* __restrict__ yn,   // (MROWS, 64)
    const float* __restrict__ Wo,   // (2048, 64) -> B[k][n] = Wo[n*64 + k]
    float* __restrict__ out)        // (MROWS, 2048)
{
    const int mt   = blockIdx.x;          // 0..511
    const int ntg  = blockIdx.y;          // 0..7  (group of 16 N tiles)
    const int wave = threadIdx.x >> 5;    // 0..3
    const int lane = threadIdx.x & 31;
    const int nt0  = ntg * 16 + wave * 4; // first of this wave's 4 N tiles

    const int row  = mt * 16 + (lane & 15);
    const int coln = lane & 15;
    const int koff = (lane >> 4) * 2;

    // load the entire A strip once: K=64 -> 16 K-chunks of 4
    const float* ap = yn + (size_t)row * DSTATE + koff;
    v2f a[16];
#pragma unroll
    for (int c = 0; c < 16; ++c)
        a[c] = *(const v2f*)(ap + 4 * c);

    const int r0 = mt * 16 + ((lane >> 4) ? 8 : 0);

#pragma unroll
    for (int nti = 0; nti < 4; ++nti) {
        const int col = (nt0 + nti) * 16 + coln;
        const float* bp = Wo + (size_t)col * DSTATE + koff;
        v8f acc = {};
#pragma unroll
        for (int c = 0; c < 16; ++c) {
            v2f bv = *(const v2f*)(bp + 4 * c);
            acc = wmma_f32(a[c], bv, acc);
        }
#pragma unroll
        for (int i = 0; i < 8; ++i)
            out[(size_t)(r0 + i) * DMODEL + col] = acc[i];
    }
}

// ---------------------------------------------------------------------------
extern "C" void kernel_launch(void* const* d_in, const int* in_sizes, int n_in,
                              void* d_out, int out_size, void* d_ws, size_t ws_size,
                              hipStream_t stream) {
    const float* x      = (const float*)d_in[0];
    const float* Wk     = (const float*)d_in[1];
    const float* Wv     = (const float*)d_in[2];
    const float* Wq     = (const float*)d_in[3];
    const float* Wa_w   = (const float*)d_in[4];
    const float* Wa_b   = (const float*)d_in[5];
    const float* lam    = (const float*)d_in[6];
    const float* norm_w = (const float*)d_in[7];
    const float* Wo     = (const float*)d_in[8];
    float*       out    = (float*)d_out;

    // workspace layout: 5 buffers of MROWS*64 f32 = 5 * 2MB = 10 MB
    const size_t R = (size_t)MROWS * DSTATE;
    float* kraw = (float*)d_ws;
    float* vraw = kraw + R;
    float* qraw = vraw + R;
    float* araw = qraw + R;
    float* ynb  = araw + R;

    // 1) projections: 512 M-tiles; 4 waves = 4 projections, 4 acc each
    proj_gemm_kernel<<<dim3(MROWS / 16), 128, 0, stream>>>(
        x, Wk, Wv, Wq, Wa_w, kraw, vraw, qraw, araw);

    // 2) l2norm(k), l2norm(q), alpha  (8 rows per 256-thread block)
    row_ops_kernel<<<dim3(MROWS / 8), 256, 0, stream>>>(
        kraw, qraw, araw, Wa_b, lam);

    // 3) sequential scan, one block per batch, fused RMSNorm
    scan_kernel<<<dim3(BATCH), 256, 0, stream>>>(
        kraw, vraw, qraw, araw, norm_w, ynb);

    // 4) output GEMM, 4 N-tiles per wave against a register-resident A strip
    out_gemm_kernel<<<dim3(MROWS / 16, DMODEL / 256), 128, 0, stream>>>(
        ynb, Wo, out);
}